// GraphEncoder_glue_635655160174
// MI455X (gfx1250) — compile-verified
//
#include <hip/hip_runtime.h>
#include <hip/hip_bf16.h>

typedef __attribute__((ext_vector_type(2))) float v2f;
typedef __attribute__((ext_vector_type(8))) float v8f;

#define NN 100000
#define DD 128

// ---------------------------------------------------------------------------
// Degree / normalization kernels
// ---------------------------------------------------------------------------
__global__ void deg_init_kernel(float* __restrict__ deg, int n) {
  int i = blockIdx.x * blockDim.x + threadIdx.x;
  if (i < n) deg[i] = 1.0f;  // self-loop contributes 1 to every node's degree
}

__global__ void deg_count_kernel(const long long* __restrict__ dst,
                                 float* __restrict__ deg, int e) {
  int i = blockIdx.x * blockDim.x + threadIdx.x;
  if (i < e) unsafeAtomicAdd(&deg[(int)dst[i]], 1.0f);
}

__global__ void dinv_kernel(const float* __restrict__ deg,
                            float* __restrict__ dinv, int n) {
  int i = blockIdx.x * blockDim.x + threadIdx.x;
  if (i < n) {
    float d = deg[i];
    dinv[i] = (d > 0.0f) ? rsqrtf(d) : 0.0f;
  }
}

// ---------------------------------------------------------------------------
// GEMM: C[N x 128] = A[N x 128] * W[128 x 128]   (fp32 WMMA 16x16x4)
// One wave per 16x16 output tile; block = 8 waves = one 16-row stripe.
// ---------------------------------------------------------------------------
__global__ __launch_bounds__(256)
void gemm_xw_kernel(const float* __restrict__ A, const float* __restrict__ W,
                    float* __restrict__ C) {
  const int lane  = threadIdx.x & 31;
  const int wid   = threadIdx.x >> 5;   // 0..7 -> column tile
  const int m     = lane & 15;
  const int hs    = lane >> 4;          // lane-half: selects K pair
  const int tileM = blockIdx.x * 16;
  const int tileN = wid * 16;

  const float* arow = A + (size_t)(tileM + m) * DD;
  v8f acc = {};
#pragma unroll
  for (int k0 = 0; k0 < DD; k0 += 4) {
    const int ka = k0 + 2 * hs;
    v2f a, b;
    a.x = arow[ka];
    a.y = arow[ka + 1];
    b.x = W[(size_t)ka * DD + tileN + m];        // B[k][n] = W[k][n]
    b.y = W[(size_t)(ka + 1) * DD + tileN + m];
    acc = __builtin_amdgcn_wmma_f32_16x16x4_f32(false, a, false, b,
                                                (short)0, acc, false, false);
  }
#pragma unroll
  for (int v = 0; v < 8; ++v) {
    const int row = tileM + v + 8 * hs;
    C[(size_t)row * DD + tileN + m] = acc[v];
  }
}

// ---------------------------------------------------------------------------
// h = conv_b + dinv^2 * h_pre  (self-loop term + bias), elementwise over N*D
// ---------------------------------------------------------------------------
__global__ void h_init_kernel(const float* __restrict__ hpre,
                              const float* __restrict__ dinv,
                              const float* __restrict__ bias,
                              float* __restrict__ h, int total) {
  int i = blockIdx.x * blockDim.x + threadIdx.x;
  if (i < total) {
    const int node = i >> 7;
    const int d    = i & (DD - 1);
    const float di = dinv[node];
    h[i] = bias[d] + di * di * hpre[i];
  }
}

// ---------------------------------------------------------------------------
// Edge scatter: h[dst] += dinv[src]*dinv[dst] * h_pre[src]
// One wave per edge; lane covers 4 contiguous dims (float4 gather, 4 atomics).
// Working set (~154 MB) fits in MI455X's 192 MB L2 -> atomics stay on-chip.
// ---------------------------------------------------------------------------
__global__ __launch_bounds__(256)
void scatter_kernel(const long long* __restrict__ src,
                    const long long* __restrict__ dst,
                    const float* __restrict__ hpre,
                    const float* __restrict__ dinv,
                    float* __restrict__ h, int e) {
  const int gwave = (int)((blockIdx.x * (size_t)blockDim.x + threadIdx.x) >> 5);
  const int lane  = threadIdx.x & 31;
  if (gwave >= e) return;
  const int s = (int)src[gwave];
  const int d = (int)dst[gwave];
  const float coef = dinv[s] * dinv[d];
  const float4 v = ((const float4*)(hpre + (size_t)s * DD))[lane];
  float* out = h + (size_t)d * DD + lane * 4;
  unsafeAtomicAdd(out + 0, coef * v.x);
  unsafeAtomicAdd(out + 1, coef * v.y);
  unsafeAtomicAdd(out + 2, coef * v.z);
  unsafeAtomicAdd(out + 3, coef * v.w);
}

// ---------------------------------------------------------------------------
// Fused heads: loc = h@loc_w.T + loc_b ; std = softplus(h@std_w.T + std_b)+eps
// z = loc + std*noise ; logvar = 2*log(std).  Two WMMA accumulators share the
// A-fragment.  B[k][n] = W[n][k] (transposed weight -> per-lane row pointer).
// ---------------------------------------------------------------------------
__global__ __launch_bounds__(256)
void head_kernel(const float* __restrict__ h,
                 const float* __restrict__ loc_w, const float* __restrict__ loc_b,
                 const float* __restrict__ std_w, const float* __restrict__ std_b,
                 const float* __restrict__ noise,
                 float* __restrict__ z_out, float* __restrict__ loc_out,
                 float* __restrict__ logvar_out) {
  const int lane  = threadIdx.x & 31;
  const int wid   = threadIdx.x >> 5;
  const int m     = lane & 15;
  const int hs    = lane >> 4;
  const int tileM = blockIdx.x * 16;
  const int tileN = wid * 16;

  const float* arow = h + (size_t)(tileM + m) * DD;
  const float* lw   = loc_w + (size_t)(tileN + m) * DD;  // row n, indexed by k
  const float* sw   = std_w + (size_t)(tileN + m) * DD;

  v8f cl = {}, cs = {};
#pragma unroll
  for (int k0 = 0; k0 < DD; k0 += 4) {
    const int ka = k0 + 2 * hs;
    v2f a, bl, bs;
    a.x  = arow[ka];  a.y  = arow[ka + 1];
    bl.x = lw[ka];    bl.y = lw[ka + 1];
    bs.x = sw[ka];    bs.y = sw[ka + 1];
    cl = __builtin_amdgcn_wmma_f32_16x16x4_f32(false, a, false, bl,
                                               (short)0, cl, false, false);
    cs = __builtin_amdgcn_wmma_f32_16x16x4_f32(false, a, false, bs,
                                               (short)0, cs, false, false);
  }

  const float lb = loc_b[tileN + m];
  const float sb = std_b[tileN + m];
#pragma unroll
  for (int v = 0; v < 8; ++v) {
    const int row = tileM + v + 8 * hs;
    const int col = tileN + m;
    const size_t idx = (size_t)row * DD + col;
    const float lp   = cl[v] + lb;
    const float spre = cs[v] + sb;
    // numerically stable softplus
    const float sp  = (spre > 0.0f) ? spre + log1pf(expf(-spre))
                                    : log1pf(expf(spre));
    const float sd  = sp + 1e-8f;
    z_out[idx]      = lp + sd * noise[idx];
    loc_out[idx]    = lp;
    logvar_out[idx] = 2.0f * logf(sd);
  }
}

// ---------------------------------------------------------------------------
extern "C" void kernel_launch(void* const* d_in, const int* in_sizes, int n_in,
                              void* d_out, int out_size, void* d_ws, size_t ws_size,
                              hipStream_t stream) {
  const long long* edge = (const long long*)d_in[0];
  const float* eps_n    = (const float*)d_in[1];
  const float* vrepr    = (const float*)d_in[2];
  const float* conv_w   = (const float*)d_in[3];
  const float* conv_b   = (const float*)d_in[4];
  const float* loc_w    = (const float*)d_in[5];
  const float* loc_b    = (const float*)d_in[6];
  const float* std_w    = (const float*)d_in[7];
  const float* std_b    = (const float*)d_in[8];

  const int E = in_sizes[0] / 2;
  const int n = in_sizes[2] / DD;          // 100000, divisible by 16

  const long long* src = edge;
  const long long* dst = edge + E;

  // workspace layout
  float* deg  = (float*)d_ws;              // n
  float* dinv = deg + n;                   // n
  float* hpre = dinv + n;                  // n*D
  float* hbuf = hpre + (size_t)n * DD;     // n*D

  float* z_out      = (float*)d_out;
  float* loc_out    = z_out + (size_t)n * DD;
  float* logvar_out = loc_out + (size_t)n * DD;

  const int T = 256;

  deg_init_kernel<<<(n + T - 1) / T, T, 0, stream>>>(deg, n);
  deg_count_kernel<<<(E + T - 1) / T, T, 0, stream>>>(dst, deg, E);
  dinv_kernel<<<(n + T - 1) / T, T, 0, stream>>>(deg, dinv, n);

  gemm_xw_kernel<<<n / 16, T, 0, stream>>>(vrepr, conv_w, hpre);

  h_init_kernel<<<((size_t)n * DD + T - 1) / T, T, 0, stream>>>(hpre, dinv,
                                                                conv_b, hbuf,
                                                                n * DD);

  // one wave (32 lanes) per edge
  const size_t sthreads = (size_t)E * 32;
  scatter_kernel<<<(sthreads + T - 1) / T, T, 0, stream>>>(src, dst, hpre,
                                                           dinv, hbuf, E);

  head_kernel<<<n / 16, T, 0, stream>>>(hbuf, loc_w, loc_b, std_w, std_b,
                                        eps_n, z_out, loc_out, logvar_out);
}